// MambaLM_62448824483848
// MI455X (gfx1250) — compile-verified
//
#include <hip/hip_runtime.h>

// ---- Model constants (match reference) ----
#define LSEQ   1024
#define DMODEL 768
#define DINNER 1536
#define DSTATE 16
#define NLAYER 4
#define NVOCAB 32000
#define XPDIM  33          // 2*DSTATE + 1
#define EPSV   1e-5f

// ---- GEMM tiling ----
#define BM 64
#define BN 128
#define BK 32
#define LDP 36             // padded LDS row stride (floats): 16B-aligned, bank-conflict-free

typedef float v2f __attribute__((ext_vector_type(2)));
typedef float v8f __attribute__((ext_vector_type(8)));

__device__ __forceinline__ float silu_f(float x) {
    return x / (1.0f + expf(-x));
}

// ---------------- Embedding gather ----------------
__global__ void k_embed(const int* __restrict__ idx,
                        const float* __restrict__ embed,
                        float* __restrict__ x) {
    int t = blockIdx.x;
    int row = idx[t];
    const float* e = embed + (size_t)row * DMODEL;
    float* xo = x + (size_t)t * DMODEL;
    for (int d = threadIdx.x; d < DMODEL; d += blockDim.x) xo[d] = e[d];
}

// ---------------- RMSNorm (one block per row) ----------------
__global__ void k_rmsnorm(const float* __restrict__ x,
                          const float* __restrict__ w,
                          float* __restrict__ o) {
    __shared__ float red[256];
    int t = blockIdx.x;
    const float* xr = x + (size_t)t * DMODEL;
    float s = 0.f;
    for (int d = threadIdx.x; d < DMODEL; d += blockDim.x) {
        float v = xr[d];
        s += v * v;
    }
    red[threadIdx.x] = s;
    __syncthreads();
    for (int st = 128; st > 0; st >>= 1) {
        if ((int)threadIdx.x < st) red[threadIdx.x] += red[threadIdx.x + st];
        __syncthreads();
    }
    float inv = rsqrtf(red[0] / (float)DMODEL + EPSV);
    float* orow = o + (size_t)t * DMODEL;
    for (int d = threadIdx.x; d < DMODEL; d += blockDim.x)
        orow[d] = xr[d] * inv * w[d];
}

// ---- async stage of one (BM x BK) A tile + (BN x BK) B tile into LDS ----
__device__ __forceinline__ void stage_tile(const float* __restrict__ A,
                                           const float* __restrict__ W,
                                           int m0, int n0, int K, int kb,
                                           unsigned int ldsA, unsigned int ldsB,
                                           int tid) {
    // A tile: BM*BK floats = 512 float4s -> 2 per thread (coalesced 128B/8 lanes)
#pragma unroll
    for (int i = 0; i < 2; ++i) {
        int idx = tid + i * 256;
        int row = idx >> 3;           // BK/4 = 8 float4 per row
        int c4  = (idx & 7) << 2;
        const float* src = A + (size_t)(m0 + row) * K + kb + c4;
        unsigned int dst = ldsA + (unsigned int)(row * LDP + c4) * 4u;
        asm volatile("global_load_async_to_lds_b128 %0, %1, off"
                     :: "v"(dst), "v"(src) : "memory");
    }
    // B tile: BN*BK floats = 1024 float4s -> 4 per thread
#pragma unroll
    for (int i = 0; i < 4; ++i) {
        int idx = tid + i * 256;
        int row = idx >> 3;
        int c4  = (idx & 7) << 2;
        const float* src = W + (size_t)(n0 + row) * K + kb + c4;
        unsigned int dst = ldsB + (unsigned int)(row * LDP + c4) * 4u;
        asm volatile("global_load_async_to_lds_b128 %0, %1, off"
                     :: "v"(dst), "v"(src) : "memory");
    }
}

// ---------------- LDS-tiled fp32 WMMA GEMM: C[M,N] = A[M,K] * W[N,K]^T ----
// Block = 256 threads (8 waves), tile BM x BN, K-blocked by BK.
// Double-buffered LDS: async loads (ASYNCcnt) for block kb+1 overlap the 32
// V_WMMA_F32_16X16X4_F32 ops on block kb; one barrier per K-step.
// Each wave computes a 32x32 output = 2x2 WMMA tiles.
// Fragment layouts per ISA: A/B lane l<16 -> {k,k+1}, l>=16 -> {k+2,k+3};
// C/D: VGPR v -> M = v + (lane>=16 ? 8:0), N = lane&15.
// Requires M%BM==0, N%BN==0, K%BK==0 (true for all GEMMs in this model).
// NT=true stores C with non-temporal hint (final logits: 131MB written once;
// keeps the 98MB embed table resident in the 192MB L2).
template <bool NT>
__global__ __launch_bounds__(256) void k_gemm_tiled(const float* __restrict__ A,
                                                    const float* __restrict__ W,
                                                    float* __restrict__ C,
                                                    int M, int N, int K) {
    __shared__ float sA[2][BM][LDP];
    __shared__ float sB[2][BN][LDP];

    const int tid  = threadIdx.x;
    const int wave = tid >> 5;
    const int lane = tid & 31;
    const int lm   = lane & 15;
    const int half2 = (lane >> 4) << 1;   // 0 or 2
    const int mw = wave >> 2;             // 0..1 -> which 32-row strip
    const int nw = wave & 3;              // 0..3 -> which 32-col strip
    const int m0 = blockIdx.y * BM;
    const int n0 = blockIdx.x * BN;

    // LDS byte offsets (flat-LDS aperture keeps the offset in addr[31:0])
    const unsigned int ldsA0 = (unsigned int)(size_t)(&sA[0][0][0]);
    const unsigned int ldsB0 = (unsigned int)(size_t)(&sB[0][0][0]);
    const unsigned int aBufB = (unsigned int)(BM * LDP * 4);
    const unsigned int bBufB = (unsigned int)(BN * LDP * 4);

    v8f acc00 = {}, acc01 = {}, acc10 = {}, acc11 = {};

    // prologue: stage first K-block into buffer 0
    stage_tile(A, W, m0, n0, K, 0, ldsA0, ldsB0, tid);

    int cur = 0;
    for (int kb = 0; kb < K; kb += BK) {
        asm volatile("s_wait_asynccnt 0x0" ::: "memory");  // this wave's stage done
        __syncthreads();                                   // all waves staged + prev compute done

        if (kb + BK < K) {   // uniform branch: overlap next stage with compute
            int nxt = cur ^ 1;
            stage_tile(A, W, m0, n0, K, kb + BK,
                       ldsA0 + (unsigned int)nxt * aBufB,
                       ldsB0 + (unsigned int)nxt * bBufB, tid);
        }

        const float (*cA)[LDP] = sA[cur];
        const float (*cB)[LDP] = sB[cur];
#pragma unroll
        for (int k = 0; k < BK; k += 4) {
            int kk = k + half2;
            v2f a0 = *(const v2f*)&cA[mw * 32 + lm][kk];
            v2f a1 = *(const v2f*)&cA[mw * 32 + 16 + lm][kk];
            v2f b0 = *(const v2f*)&cB[nw * 32 + lm][kk];
            v2f b1 = *(const v2f*)&cB[nw * 32 + 16 + lm][kk];
            acc00 = __builtin_amdgcn_wmma_f32_16x16x4_f32(false, a0, false, b0, (short)0, acc00, false, false);
            acc01 = __builtin_amdgcn_wmma_f32_16x16x4_f32(false, a0, false, b1, (short)0, acc01, false, false);
            acc10 = __builtin_amdgcn_wmma_f32_16x16x4_f32(false, a1, false, b0, (short)0, acc10, false, false);
            acc11 = __builtin_amdgcn_wmma_f32_16x16x4_f32(false, a1, false, b1, (short)0, acc11, false, false);
        }
        cur ^= 1;
    }

    const int moff = (lane >> 4) << 3;    // 0 or 8
    v8f accs[2][2] = { { acc00, acc01 }, { acc10, acc11 } };
#pragma unroll
    for (int mi = 0; mi < 2; ++mi) {
#pragma unroll
        for (int ni = 0; ni < 2; ++ni) {
            float* cb = C + (size_t)(m0 + mw * 32 + mi * 16 + moff) * N
                          + (n0 + nw * 32 + ni * 16 + lm);
            v8f a = accs[mi][ni];
#pragma unroll
            for (int v = 0; v < 8; ++v) {
                if (NT) __builtin_nontemporal_store(a[v], &cb[(size_t)v * N]);
                else    cb[(size_t)v * N] = a[v];
            }
        }
    }
}

// ---------------- Depthwise causal conv (width 4) + SiLU ----------------
__global__ void k_conv(const float* __restrict__ xz,
                       const float* __restrict__ cw,
                       const float* __restrict__ cb,
                       float* __restrict__ xc) {
    int i = blockIdx.x * blockDim.x + threadIdx.x;  // over LSEQ*DINNER
    int d = i % DINNER;
    int t = i / DINNER;
    const float* w = cw + (size_t)d * 4;
    float acc = cb[d];
#pragma unroll
    for (int j = 0; j < 4; ++j) {
        int tt = t - 3 + j;
        if (tt >= 0) acc += w[j] * xz[(size_t)tt * (2 * DINNER) + d];
    }
    xc[(size_t)t * DINNER + d] = silu_f(acc);
}

// ---------------- xp = xc @ W_x^T (N = 33, tiny) ----------------
__global__ void k_xp(const float* __restrict__ xc,
                     const float* __restrict__ Wx,
                     float* __restrict__ xp) {
    int t = blockIdx.x;
    int j = threadIdx.x;
    if (j >= XPDIM) return;
    const float* xr = xc + (size_t)t * DINNER;
    const float* wr = Wx + (size_t)j * DINNER;
    float s = 0.f;
    for (int k = 0; k < DINNER; ++k) s += xr[k] * wr[k];
    xp[t * XPDIM + j] = s;
}

// ---------------- Selective scan: one thread per channel d ----------------
__global__ void k_scan(const float* __restrict__ xp,
                       const float* __restrict__ xc,
                       const float* __restrict__ A_log,
                       const float* __restrict__ dt_w,
                       const float* __restrict__ dt_b,
                       const float* __restrict__ Dp,
                       float* __restrict__ y) {
    __shared__ float sxp[XPDIM];
    int d = blockIdx.x * blockDim.x + threadIdx.x;

    float A[DSTATE];
#pragma unroll
    for (int n = 0; n < DSTATE; ++n) A[n] = -expf(A_log[(size_t)d * DSTATE + n]);
    float dw = dt_w[d], db = dt_b[d], Dpd = Dp[d];

    float h[DSTATE];
#pragma unroll
    for (int n = 0; n < DSTATE; ++n) h[n] = 0.f;

    for (int t = 0; t < LSEQ; ++t) {
        if ((int)threadIdx.x < XPDIM) sxp[threadIdx.x] = xp[t * XPDIM + threadIdx.x];
        __syncthreads();

        float dtr = sxp[0];
        float xv = xc[(size_t)t * DINNER + d];
        float u = dtr * dw + db;
        float dt = (u > 20.f) ? u : log1pf(expf(u));   // softplus

        float acc = 0.f;
#pragma unroll
        for (int n = 0; n < DSTATE; ++n) {
            float dA = expf(dt * A[n]);
            h[n] = dA * h[n] + dt * sxp[1 + n] * xv;
            acc += h[n] * sxp[1 + DSTATE + n];
        }
        y[(size_t)t * DINNER + d] = acc + xv * Dpd;
        __syncthreads();
    }
}

// ---------------- y *= silu(z) ----------------
__global__ void k_gate(float* __restrict__ y, const float* __restrict__ xz) {
    int i = blockIdx.x * blockDim.x + threadIdx.x;
    int d = i % DINNER;
    int t = i / DINNER;
    float z = xz[(size_t)t * (2 * DINNER) + DINNER + d];
    y[i] *= silu_f(z);
}

// ---------------- residual add ----------------
__global__ void k_add(float* __restrict__ x, const float* __restrict__ r) {
    int i = blockIdx.x * blockDim.x + threadIdx.x;
    x[i] += r[i];
}

extern "C" void kernel_launch(void* const* d_in, const int* in_sizes, int n_in,
                              void* d_out, int out_size, void* d_ws, size_t ws_size,
                              hipStream_t stream) {
    (void)in_sizes; (void)n_in; (void)out_size; (void)ws_size;

    const int*   idx      = (const int*)d_in[0];
    const float* embed    = (const float*)d_in[1];
    const float* norm_w   = (const float*)d_in[2];
    const float* W_in     = (const float*)d_in[3];
    const float* conv_w   = (const float*)d_in[4];
    const float* conv_b   = (const float*)d_in[5];
    const float* W_x      = (const float*)d_in[6];
    const float* dt_w     = (const float*)d_in[7];
    const float* dt_b     = (const float*)d_in[8];
    const float* A_log    = (const float*)d_in[9];
    const float* Dp       = (const float*)d_in[10];
    const float* out_w    = (const float*)d_in[11];
    const float* norm_f_w = (const float*)d_in[12];

    float* ws = (float*)d_ws;
    size_t off = 0;
    float* x   = ws + off; off += (size_t)LSEQ * DMODEL;        // residual stream
    float* xn  = ws + off; off += (size_t)LSEQ * DMODEL;        // normalized
    float* xz  = ws + off; off += (size_t)LSEQ * 2 * DINNER;    // in-proj
    float* xc  = ws + off; off += (size_t)LSEQ * DINNER;        // conv+silu
    float* xpb = ws + off; off += (size_t)LSEQ * XPDIM;         // dt/B/C proj
    float* y   = ws + off; off += (size_t)LSEQ * DINNER;        // scan output
    float* tmp = ws + off; off += (size_t)LSEQ * DMODEL;        // layer output

    k_embed<<<LSEQ, 256, 0, stream>>>(idx, embed, x);

    for (int i = 0; i < NLAYER; ++i) {
        const float* Wi   = W_in   + (size_t)i * 2 * DINNER * DMODEL;
        const float* cwi  = conv_w + (size_t)i * DINNER * 4;
        const float* cbi  = conv_b + (size_t)i * DINNER;
        const float* Wxi  = W_x    + (size_t)i * XPDIM * DINNER;
        const float* dwi  = dt_w   + (size_t)i * DINNER;
        const float* dbi  = dt_b   + (size_t)i * DINNER;
        const float* Ali  = A_log  + (size_t)i * DINNER * DSTATE;
        const float* Dpi  = Dp     + (size_t)i * DINNER;
        const float* Owi  = out_w  + (size_t)i * DMODEL * DINNER;
        const float* nwi  = norm_w + (size_t)i * DMODEL;

        k_rmsnorm<<<LSEQ, 256, 0, stream>>>(x, nwi, xn);

        dim3 g1((2 * DINNER) / BN, LSEQ / BM);
        k_gemm_tiled<false><<<g1, 256, 0, stream>>>(xn, Wi, xz, LSEQ, 2 * DINNER, DMODEL);

        k_conv<<<(LSEQ * DINNER) / 256, 256, 0, stream>>>(xz, cwi, cbi, xc);

        k_xp<<<LSEQ, 64, 0, stream>>>(xc, Wxi, xpb);

        k_scan<<<DINNER / 256, 256, 0, stream>>>(xpb, xc, Ali, dwi, dbi, Dpi, y);

        k_gate<<<(LSEQ * DINNER) / 256, 256, 0, stream>>>(y, xz);

        dim3 g2(DMODEL / BN, LSEQ / BM);
        k_gemm_tiled<false><<<g2, 256, 0, stream>>>(y, Owi, tmp, LSEQ, DMODEL, DINNER);

        k_add<<<(LSEQ * DMODEL) / 256, 256, 0, stream>>>(x, tmp);
    }

    k_rmsnorm<<<LSEQ, 256, 0, stream>>>(x, norm_f_w, xn);

    dim3 g3(NVOCAB / BN, LSEQ / BM);
    k_gemm_tiled<true><<<g3, 256, 0, stream>>>(xn, embed, (float*)d_out, LSEQ, NVOCAB, DMODEL);
}